// AdaConv2D_56057913147427
// MI455X (gfx1250) — compile-verified
//
#include <hip/hip_runtime.h>
#include <math.h>

typedef __attribute__((ext_vector_type(2))) float v2f;
typedef __attribute__((ext_vector_type(4))) float v4f;
typedef __attribute__((ext_vector_type(8))) float v8f;

#define B_    8
#define C_    512
#define H_    128
#define W_    128
#define G_    128          // groups per batch
#define CPG   4            // channels per group
#define KTAPS 9
#define NCH   (B_ * C_)    // 4096 (b,c) channels
#define HW    (H_ * W_)    // 16384 pixels

#define AS1 __attribute__((address_space(1)))
#define AS3 __attribute__((address_space(3)))

#if defined(__has_builtin)
#if __has_builtin(__builtin_amdgcn_global_load_async_to_lds_b32)
#define HAVE_ASYNC_LDS 1
#endif
#endif

// ---------------------------------------------------------------------------
// Kernel 1: per-channel mean and 1/(std+eps)  (unbiased std, ddof=1)
// ---------------------------------------------------------------------------
__global__ __launch_bounds__(256) void stats_kernel(const float* __restrict__ x,
                                                    float* __restrict__ meanOut,
                                                    float* __restrict__ rstdOut) {
    const int ch = blockIdx.x;                       // 0..4095
    const float* p = x + (size_t)ch * HW;
    float s = 0.f, s2 = 0.f;
    for (int i = threadIdx.x * 4; i < HW; i += 256 * 4) {
        v4f v = *(const v4f*)(p + i);
        s  += v.x + v.y + v.z + v.w;
        s2 += v.x * v.x + v.y * v.y + v.z * v.z + v.w * v.w;
    }
    __shared__ float ls[256];
    __shared__ float ls2[256];
    ls[threadIdx.x] = s;
    ls2[threadIdx.x] = s2;
    __syncthreads();
    for (int off = 128; off > 0; off >>= 1) {
        if ((int)threadIdx.x < off) {
            ls[threadIdx.x]  += ls[threadIdx.x + off];
            ls2[threadIdx.x] += ls2[threadIdx.x + off];
        }
        __syncthreads();
    }
    if (threadIdx.x == 0) {
        float m   = ls[0] / (float)HW;
        float var = (ls2[0] - (float)HW * m * m) / (float)(HW - 1);
        var = fmaxf(var, 0.f);
        float sd = sqrtf(var) + 1e-7f;
        meanOut[ch] = m;
        rstdOut[ch] = 1.f / sd;
    }
}

// ---------------------------------------------------------------------------
// Kernel 2: compose pw∘dw and fold rstd:
//   wfin[b][g][oc][ic][t] = rstd[b,g*4+ic] * Σ_mc pw[b,g*4+oc][mc] * dw[b,g*4+mc][ic][t]
// ---------------------------------------------------------------------------
__global__ __launch_bounds__(256) void compose_kernel(const float* __restrict__ dw,
                                                      const float* __restrict__ pw,
                                                      const float* __restrict__ rstd,
                                                      float* __restrict__ wfin) {
    int idx = blockIdx.x * 256 + threadIdx.x;
    if (idx >= NCH * CPG * KTAPS) return;            // 4096*36
    int t = idx % KTAPS;  int r = idx / KTAPS;
    int ic = r & 3;       r >>= 2;
    int oc = r & 3;       r >>= 2;
    int g  = r % G_;      int b = r / G_;
    float acc = 0.f;
    for (int mc = 0; mc < 4; ++mc) {
        float wdw = dw[(((size_t)b * C_ + g * 4 + mc) * CPG + ic) * KTAPS + t];
        float wpw = pw[((size_t)b * C_ + g * 4 + oc) * CPG + mc];
        acc += wdw * wpw;
    }
    wfin[idx] = acc * rstd[b * C_ + g * 4 + ic];
}

// ---------------------------------------------------------------------------
// Kernel 3: effective bias:  beff = bias - Σ_{ic,t} wfin * mean[ic]
// ---------------------------------------------------------------------------
__global__ __launch_bounds__(256) void bias_kernel(const float* __restrict__ biases,
                                                   const float* __restrict__ wfin,
                                                   const float* __restrict__ meanIn,
                                                   float* __restrict__ beff) {
    int idx = blockIdx.x * 256 + threadIdx.x;        // = b*512 + g*4 + oc
    if (idx >= NCH) return;
    int g  = (idx >> 2) % G_;
    int b  = idx / C_;
    float acc = biases[idx];
    const float* wp = wfin + (size_t)idx * (CPG * KTAPS);
    for (int ic = 0; ic < 4; ++ic) {
        float m = meanIn[b * C_ + g * 4 + ic];
        for (int t = 0; t < KTAPS; ++t) acc -= wp[ic * KTAPS + t] * m;
    }
    beff[idx] = acc;
}

// ---------------------------------------------------------------------------
// Kernel 4: fused grouped conv via V_WMMA_F32_16X16X4_F32.
// Block = 256 thr (8 waves). Grid = (rowtile 0..7, group 0..127, batch 0..7).
//
// LDS layout (ic-pair interleaved so each A fragment is ONE aligned b64 load):
//   element (h, r, c, p) at float offset  h*PPSTRIDE + (r*132 + c)*2 + p
//   h = ic>>1 (lane-half / K-base), p = ic&1, r = 0..17 rows, c = 0..131 cols.
//   PPSTRIDE = 18*132*2 + 16  => pair delta == 32 mod 64 banks: the two lane
//   halves of a wave hit disjoint bank ranges -> conflict-free ds_load_b64.
//
// Per wave sub-tile: M=16 x-pixels, N=16 (4 valid oc), K=4 ic; 9 WMMAs (taps).
// ---------------------------------------------------------------------------
#define XT_W 132
#define XT_H 18
#define PPSTRIDE (XT_H * XT_W * 2 + 16)              // 4768 floats
#define STAGE_N  (2 * XT_H * 2 * XT_W)               // 9504 cells

__global__ __launch_bounds__(256) void conv_kernel(const float* __restrict__ x,
                                                   const float* __restrict__ wfin,
                                                   const float* __restrict__ beff,
                                                   float* __restrict__ out) {
    __shared__ float xs[2 * PPSTRIDE];               // 38144 B

    const int rt = blockIdx.x;                       // row tile (16 rows)
    const int g  = blockIdx.y;
    const int b  = blockIdx.z;
    const int y0 = rt * 16;

    const float* xbase = x + (size_t)(b * C_ + g * 4) * HW;

    // ---- stage input tile (zero-padded halo) -------------------------------
    for (int idx = threadIdx.x; idx < STAGE_N; idx += 256) {
        int c  = idx % XT_W;
        int t2 = idx / XT_W;                         // (h*18 + r)*2 + p
        int p  = t2 & 1;
        int hr = t2 >> 1;
        int r  = hr % XT_H;
        int h  = hr / XT_H;
        int ic = h * 2 + p;
        int gy = y0 - 1 + r;
        int gx = c - 1;
        int loff = h * PPSTRIDE + (r * XT_W + c) * 2 + p;
        bool valid = (gy >= 0) & (gy < H_) & (gx >= 0) & (gx < W_);
#if HAVE_ASYNC_LDS
        if (valid) {
            // direct global -> LDS, tracked by ASYNCcnt (no VGPR round-trip)
            AS1 int* gp =
                (AS1 int*)(uintptr_t)(xbase + (size_t)ic * HW + gy * W_ + gx);
            AS3 int* lp =
                (AS3 int*)(unsigned)(uintptr_t)(&xs[loff]);   // ISA §10.2: flat LDS addr[31:0] = LDS offset
            __builtin_amdgcn_global_load_async_to_lds_b32(gp, lp, 0, 0);
        } else {
            xs[loff] = 0.f;
        }
#else
        float v = 0.f;
        if (valid) v = xbase[(size_t)ic * HW + gy * W_ + gx];
        xs[loff] = v;
#endif
    }

    // ---- per-lane B fragments (ISA 32-bit B 4x16 layout) + fused bias ------
    const int lane = threadIdx.x & 31;
    const int n    = lane & 15;                      // N column (oc when <4)
    const int h    = lane >> 4;                      // plane pair 0/1
    const int kb   = h << 1;                         // K base: 0 or 2
    const float* wp = wfin + (size_t)(b * G_ + g) * (CPG * CPG * KTAPS);
    v2f bw[KTAPS];
    #pragma unroll
    for (int t = 0; t < KTAPS; ++t) {
        float b0 = 0.f, b1 = 0.f;
        if (n < 4) {
            b0 = wp[(n * CPG + kb + 0) * KTAPS + t];
            b1 = wp[(n * CPG + kb + 1) * KTAPS + t];
        }
        bw[t].x = b0;
        bw[t].y = b1;
    }
    const float biasv = (n < 4) ? beff[(b * G_ + g) * 4 + n] : 0.f;

#if HAVE_ASYNC_LDS
#if defined(__has_builtin) && __has_builtin(__builtin_amdgcn_s_wait_asynccnt)
    __builtin_amdgcn_s_wait_asynccnt(0);
#else
    asm volatile("s_wait_asynccnt 0x0" ::: "memory");
#endif
#endif
    __syncthreads();

    const int wave = threadIdx.x >> 5;               // 0..7
    float* obase = out + (size_t)(b * C_ + g * 4) * HW;
    const int abase = h * PPSTRIDE;                  // lane-half plane select

    // 128 sub-tiles: 16 rows x 8 x-tiles, round-robined over 8 waves
    for (int tIdx = wave; tIdx < 128; tIdx += 8) {
        const int row = tIdx >> 3;                   // 0..15
        const int x0  = (tIdx & 7) * 16;

        v8f acc = {biasv, biasv, biasv, biasv, biasv, biasv, biasv, biasv};

        #pragma unroll
        for (int ky = 0; ky < 3; ++ky) {
            #pragma unroll
            for (int kx = 0; kx < 3; ++kx) {
                const int t = ky * 3 + kx;
                // A fragment: one aligned ds_load_b64 (K-pair for this lane half)
                v2f a = *(const v2f*)&xs[abase + ((row + ky) * XT_W + (x0 + n + kx)) * 2];
                acc = __builtin_amdgcn_wmma_f32_16x16x4_f32(
                    /*neg_a=*/false, a, /*neg_b=*/false, bw[t],
                    /*c_mod=*/(short)0, acc, /*reuse_a=*/false, /*reuse_b=*/false);
            }
        }

        // D layout: lane (n, half) holds M = mbase..mbase+7 for column n
        if (n < 4) {
            const int mbase = h * 8;
            float* op = obase + (size_t)n * HW + (size_t)(y0 + row) * W_ + x0 + mbase;
            v4f lo = {acc[0], acc[1], acc[2], acc[3]};
            v4f hi = {acc[4], acc[5], acc[6], acc[7]};
            *(v4f*)op       = lo;
            *(v4f*)(op + 4) = hi;
        }
    }
}

// ---------------------------------------------------------------------------
// Launcher
// ---------------------------------------------------------------------------
extern "C" void kernel_launch(void* const* d_in, const int* in_sizes, int n_in,
                              void* d_out, int out_size, void* d_ws, size_t ws_size,
                              hipStream_t stream) {
    const float* x   = (const float*)d_in[0];
    const float* dw  = (const float*)d_in[1];
    const float* pw  = (const float*)d_in[2];
    const float* bia = (const float*)d_in[3];
    float* out = (float*)d_out;

    float* mean = (float*)d_ws;                       //   4096 f
    float* rstd = mean + NCH;                         //   4096 f
    float* wfin = rstd + NCH;                         // 147456 f
    float* beff = wfin + (size_t)NCH * CPG * KTAPS;   //   4096 f  (total ~640 KB)

    stats_kernel<<<NCH, 256, 0, stream>>>(x, mean, rstd);

    const int nW = NCH * CPG * KTAPS;                 // 147456
    compose_kernel<<<(nW + 255) / 256, 256, 0, stream>>>(dw, pw, rstd, wfin);

    bias_kernel<<<(NCH + 255) / 256, 256, 0, stream>>>(bia, wfin, mean, beff);

    dim3 grid(H_ / 16, G_, B_);                       // (8, 128, 8)
    conv_kernel<<<grid, 256, 0, stream>>>(x, wfin, beff, out);
}